// ConvGNN_12661563588917
// MI455X (gfx1250) — compile-verified
//
#include <hip/hip_runtime.h>

typedef float v2f __attribute__((ext_vector_type(2)));
typedef float v8f __attribute__((ext_vector_type(8)));

constexpr int NPTS = 50000;
constexpr int KNBR = 16;
constexpr int CIN  = 64;
constexpr int COUT = 64;
constexpr int NB   = 4;
constexpr int NTILES = NPTS / 16;            // 3125 (exact)
constexpr int WPB  = 8;                      // waves per block (wave32)
constexpr int TPB  = WPB * 32;               // 256 threads
constexpr int LDSW_FLOATS = KNBR * 16 * 2 * 64 * 2; // 65536 floats = 256 KB

// One wave computes out[0..3][n0..n0+15][0..63] via V_WMMA_F32_16X16X4_F32.
// A-fragments are gathered directly from global (L2-resident batch) in WMMA
// register layout; B-fragments come from LDS pre-swizzled weights.
__global__ __launch_bounds__(TPB)
void conv_gnn_wmma_f32(const float* __restrict__ xin,
                       const int*   __restrict__ nbr,
                       const float* __restrict__ wgt,
                       const float* __restrict__ bias,
                       float*       __restrict__ out)
{
  extern __shared__ float ldsw[];  // 256 KB weight stage
  const int tid = threadIdx.x;

  // Stage weights into B-fragment order:
  // pair p -> (k, cb, half, o): ldsw[2p+0] = W[k][cb*4+half*2+0][o]
  //                             ldsw[2p+1] = W[k][cb*4+half*2+1][o]
  for (int p = tid; p < LDSW_FLOATS / 2; p += TPB) {
    const int o    = p & 63;
    const int half = (p >> 6) & 1;
    const int cb   = (p >> 7) & 15;
    const int k    = p >> 11;
    const int c0   = cb * 4 + half * 2;
    const int src  = (k * CIN + c0) * COUT + o;
    ldsw[2 * p]     = wgt[src];
    ldsw[2 * p + 1] = wgt[src + COUT];
  }
  __syncthreads();

  const int wave = tid >> 5;
  const int lane = tid & 31;
  const int tile = blockIdx.x * WPB + wave;
  if (tile >= NTILES) return;          // whole-wave exit: EXEC stays all-1s for WMMA

  const int n0   = tile * 16;
  const int m    = lane & 15;          // A row / B,D column within the 16x16 tile
  const int half = lane >> 4;          // selects K-pair within K=4 step

  // Per-lane neighbor byte offsets for row (n0+m); negative sentinel => zero row.
  int roff[KNBR];
#pragma unroll
  for (int k = 0; k < KNBR; ++k) {
    const int idx = nbr[(n0 + m) * KNBR + k];
    roff[k] = (idx < NPTS) ? idx * (CIN * 4) : -1;
  }

  const char* bb[NB];
#pragma unroll
  for (int b = 0; b < NB; ++b)
    bb[b] = (const char*)xin + (size_t)b * (size_t)NPTS * CIN * 4 + half * 8;

  v8f acc[NB][4];
#pragma unroll
  for (int b = 0; b < NB; ++b)
#pragma unroll
    for (int ot = 0; ot < 4; ++ot)
      acc[b][ot] = {};

#pragma unroll
  for (int k = 0; k < KNBR; ++k) {
    const bool valid = roff[k] >= 0;
    const int  ro    = valid ? roff[k] : 0;
    // Base of this k's B-fragment region in LDS (floats):
    const float* lw = &ldsw[(k * 32 + half) * 128 + m * 2];
#pragma unroll 4
    for (int cb = 0; cb < 16; ++cb) {
      v2f a[NB];
#pragma unroll
      for (int b = 0; b < NB; ++b) {
        v2f t = *(const v2f*)(bb[b] + ro + cb * 16);
        a[b] = valid ? t : v2f{0.0f, 0.0f};
      }
      v2f bw[4];
#pragma unroll
      for (int ot = 0; ot < 4; ++ot)
        bw[ot] = *(const v2f*)(lw + cb * 256 + ot * 32);
#pragma unroll
      for (int b = 0; b < NB; ++b)
#pragma unroll
        for (int ot = 0; ot < 4; ++ot)
          acc[b][ot] = __builtin_amdgcn_wmma_f32_16x16x4_f32(
              false, a[b], false, bw[ot], (short)0, acc[b][ot], false, false);
    }
  }

  // D layout: VGPR j -> rows (j, j+8) for lane halves; column = m within cout-tile.
#pragma unroll
  for (int b = 0; b < NB; ++b) {
#pragma unroll
    for (int ot = 0; ot < 4; ++ot) {
      const float bs = bias[ot * 16 + m];
#pragma unroll
      for (int j = 0; j < 8; ++j) {
        const int row = n0 + half * 8 + j;
        out[((size_t)b * NPTS + row) * COUT + ot * 16 + m] = acc[b][ot][j] + bs;
      }
    }
  }
}

extern "C" void kernel_launch(void* const* d_in, const int* in_sizes, int n_in,
                              void* d_out, int out_size, void* d_ws, size_t ws_size,
                              hipStream_t stream) {
  (void)in_sizes; (void)n_in; (void)d_ws; (void)ws_size; (void)out_size;
  const float* xin  = (const float*)d_in[0];
  const int*   nbr  = (const int*)d_in[1];
  const float* wgt  = (const float*)d_in[2];
  const float* bias = (const float*)d_in[3];
  float*       out  = (float*)d_out;

  const size_t shmem = (size_t)LDSW_FLOATS * sizeof(float); // 262144 bytes
  hipFuncSetAttribute((const void*)conv_gnn_wmma_f32,
                      hipFuncAttributeMaxDynamicSharedMemorySize, (int)shmem);

  const int blocks = (NTILES + WPB - 1) / WPB;   // 391
  conv_gnn_wmma_f32<<<blocks, TPB, shmem, stream>>>(xin, nbr, wgt, bias, out);
}